// HybridGNN_34548716929462
// MI455X (gfx1250) — compile-verified
//
#include <hip/hip_runtime.h>
#include <hip/hip_bf16.h>

// ---------------------------------------------------------------------------
// HybridGNN for MI455X (gfx1250, wave32).
// Dense GEMMs use V_WMMA_F32_16X16X4_F32: one wave per 16x16 output tile,
// 8 waves per 256-thread block covering H=128 columns. Activation panels are
// staged once per block into LDS (coalesced float4 loads, padded pitch so
// A-fragment ds_load_b64 reads are bank-conflict-free); the mean-divide is
// folded into staging. Scatter stages use f32 atomics absorbed by 192MB L2.
// ---------------------------------------------------------------------------

typedef __attribute__((ext_vector_type(2))) float v2f;
typedef __attribute__((ext_vector_type(8))) float v8f;

#define NUsr 100000
#define NMer 20000
#define FUin 64
#define FMin 32
#define HH   128
#define NEdg 1000000
#define NTgt 500000
#define PITCH 260   // 256 + 4 floats: row r, col k -> bank (4r + k) % 64

// ---------------- zero the accumulator region ------------------------------
__global__ void zero_kernel(float4* __restrict__ p, long n4) {
  long i = (long)blockIdx.x * blockDim.x + threadIdx.x;
  if (i < n4) p[i] = make_float4(0.f, 0.f, 0.f, 0.f);
}

// ---------------- degree count ---------------------------------------------
__global__ void degree_kernel(const int* __restrict__ col, float* __restrict__ deg, int E) {
  int e = blockIdx.x * blockDim.x + threadIdx.x;
  if (e < E) atomicAdd(&deg[col[e]], 1.0f);
}

// ---------------- edge scatter-add: agg[col] += x[row] ---------------------
__global__ void scatter_kernel(const float* __restrict__ x, const int* __restrict__ row,
                               const int* __restrict__ col, float* __restrict__ agg,
                               int E, int F) {
  int per = F >> 2;
  long t = (long)blockIdx.x * blockDim.x + threadIdx.x;
  int e = (int)(t / per);
  int f = (int)(t % per) << 2;
  if (e >= E) return;
  const float4 v = *(const float4*)(x + (size_t)row[e] * F + f);
  float* dst = agg + (size_t)col[e] * F + f;
  atomicAdd(dst + 0, v.x);
  atomicAdd(dst + 1, v.y);
  atomicAdd(dst + 2, v.z);
  atomicAdd(dst + 3, v.w);
}

// ---------------- fused SAGE layer: act((agg/deg)@Wl + xdst@Wr + b) --------
// 16-row tile per block, LDS-staged [16 x (K1+K2)] activation panel with the
// 1/deg scale applied during staging. 8 waves cover the 8 column tiles of H.
// WMMA f32 16x16x4 fragment layout (ISA 7.12.2):
//   A (16x4): lanes 0-15 -> M=lane, VGPR0/1 = K=0/1; lanes 16-31 -> K=2/3.
//   B (4x16): lanes 0-15 -> N=lane, VGPR0/1 = K=0/1; lanes 16-31 -> K=2/3.
//   C/D     : VGPR j, lane L -> row = j + (L>=16 ? 8 : 0), col = L%16.
template <int K1, int K2, int DORELU>
__global__ __launch_bounds__(256)
void sage_gemm_kernel(const float* __restrict__ agg, const float* __restrict__ deg,
                      const float* __restrict__ xdst,
                      const float* __restrict__ Wl, const float* __restrict__ Wr,
                      const float* __restrict__ bias, float* __restrict__ out) {
  __shared__ float sA[16 * PITCH];
  const int tid   = threadIdx.x;
  const int mbase = blockIdx.x * 16;

  // Cooperative coalesced staging: cols [0,K1) = agg * 1/max(deg,1); [K1,K1+K2) = xdst.
  constexpr int C1 = (16 * K1) >> 2;          // float4 chunks in agg region
  constexpr int CT = (16 * (K1 + K2)) >> 2;   // total float4 chunks
#pragma unroll
  for (int c = tid; c < CT; c += 256) {
    if (c < C1) {
      const int r = c / (K1 >> 2);
      const int f = (c % (K1 >> 2)) << 2;
      const float invd = 1.0f / fmaxf(deg[mbase + r], 1.0f);
      const float4 v = *(const float4*)(agg + (size_t)(mbase + r) * K1 + f);
      float* d = sA + r * PITCH + f;
      d[0] = v.x * invd; d[1] = v.y * invd; d[2] = v.z * invd; d[3] = v.w * invd;
    } else {
      const int c2 = c - C1;
      const int r = c2 / (K2 >> 2);
      const int f = (c2 % (K2 >> 2)) << 2;
      const float4 v = *(const float4*)(xdst + (size_t)(mbase + r) * K2 + f);
      float* d = sA + r * PITCH + K1 + f;
      d[0] = v.x; d[1] = v.y; d[2] = v.z; d[3] = v.w;
    }
  }
  __syncthreads();

  const int lane = tid & 31;
  const int wave = tid >> 5;
  const int half = lane >> 4;
  const int l16  = lane & 15;
  const int ncol = wave * 16 + l16;
  const float* sArow = sA + l16 * PITCH;

  v8f c = {};
#pragma unroll
  for (int k = 0; k < K1 + K2; k += 4) {
    const int ka = k + half * 2;
    const float* wsrc = (k < K1) ? (Wl + (size_t)(k + half * 2) * HH)
                                 : (Wr + (size_t)(k - K1 + half * 2) * HH);
    v2f a, b;
    a.x = sArow[ka];            // ds_load_b64, conflict-free via padded pitch
    a.y = sArow[ka + 1];
    b.x = wsrc[ncol];
    b.y = wsrc[HH + ncol];
    c = __builtin_amdgcn_wmma_f32_16x16x4_f32(false, a, false, b, (short)0, c, false, false);
  }

  const float bv = bias[ncol];
#pragma unroll
  for (int j = 0; j < 8; ++j) {
    float v = c[j] + bv;
    if (DORELU) v = fmaxf(v, 0.0f);
    out[(size_t)(mbase + half * 8 + j) * HH + ncol] = v;
  }
}

// ---------------- edge classifier: relu([zu|zm]@Wc1+bc1)@Wc2+bc2 -----------
// 16 target edges per block. The per-edge gather of z_u[row_t]/z_m[col_t] is
// staged as a coalesced 16x256 LDS panel (each source row is 128 contiguous
// floats). Each wave computes one 16x16 hidden tile, folds in its Wc2 column,
// shuffle-reduces across its 16 columns, and the 8 waves combine via LDS.
__global__ __launch_bounds__(256)
void edge_mlp_kernel(const float* __restrict__ zu, const float* __restrict__ zm,
                     const int* __restrict__ rowt, const int* __restrict__ colt,
                     const float* __restrict__ Wc1, const float* __restrict__ bc1,
                     const float* __restrict__ Wc2, const float* __restrict__ bc2,
                     float* __restrict__ out) {
  __shared__ float sZ[16 * PITCH];
  __shared__ float s_out[16];
  const int tid   = threadIdx.x;
  const int ebase = blockIdx.x * 16;

  // Stage: row r cols [0,128) = zu[rowt[e]], cols [128,256) = zm[colt[e]].
#pragma unroll
  for (int c = tid; c < 16 * 64; c += 256) {  // 64 float4 chunks per row
    const int r = c >> 6;
    const int q = c & 63;
    const int e = ebase + r;
    const float* src = (q < 32) ? (zu + (size_t)rowt[e] * HH + (q << 2))
                                : (zm + (size_t)colt[e] * HH + ((q - 32) << 2));
    const float4 v = *(const float4*)src;
    float* d = sZ + r * PITCH + (q << 2);
    d[0] = v.x; d[1] = v.y; d[2] = v.z; d[3] = v.w;
  }
  if (tid < 16) s_out[tid] = 0.0f;
  __syncthreads();

  const int lane = tid & 31;
  const int wave = tid >> 5;
  const int half = lane >> 4;
  const int l16  = lane & 15;
  const int ncol = wave * 16 + l16;
  const float* sZrow = sZ + l16 * PITCH;

  v8f c = {};
#pragma unroll
  for (int k = 0; k < 2 * HH; k += 4) {
    const int ka = k + half * 2;
    v2f a, b;
    a.x = sZrow[ka];
    a.y = sZrow[ka + 1];
    b.x = Wc1[(size_t)ka * HH + ncol];
    b.y = Wc1[(size_t)(ka + 1) * HH + ncol];
    c = __builtin_amdgcn_wmma_f32_16x16x4_f32(false, a, false, b, (short)0, c, false, false);
  }

  const float bv = bc1[ncol];
  const float w2 = Wc2[ncol];

  float part[8];
#pragma unroll
  for (int j = 0; j < 8; ++j)
    part[j] = fmaxf(c[j] + bv, 0.0f) * w2;

  // butterfly reduce across the 16 lanes of each half (same rows, diff cols)
#pragma unroll
  for (int m = 1; m <= 8; m <<= 1) {
#pragma unroll
    for (int j = 0; j < 8; ++j)
      part[j] += __shfl_xor(part[j], m, 32);
  }
  if (l16 == 0) {
#pragma unroll
    for (int j = 0; j < 8; ++j)
      atomicAdd(&s_out[half * 8 + j], part[j]);   // ds_add_f32
  }
  __syncthreads();

  if (tid < 16)
    out[(size_t)ebase + tid] = s_out[tid] + bc2[0];
}

// ---------------------------------------------------------------------------
extern "C" void kernel_launch(void* const* d_in, const int* in_sizes, int n_in,
                              void* d_out, int out_size, void* d_ws, size_t ws_size,
                              hipStream_t stream) {
  (void)in_sizes; (void)n_in; (void)out_size; (void)ws_size;

  const float* x_user  = (const float*)d_in[0];
  const float* x_merch = (const float*)d_in[1];
  const int*   row_um  = (const int*)d_in[2];
  const int*   col_um  = (const int*)d_in[3];
  const int*   row_mu  = (const int*)d_in[4];
  const int*   col_mu  = (const int*)d_in[5];
  const int*   row_t   = (const int*)d_in[6];
  const int*   col_t   = (const int*)d_in[7];
  const float* W1l_um  = (const float*)d_in[8];
  const float* W1r_um  = (const float*)d_in[9];
  const float* b1_um   = (const float*)d_in[10];
  const float* W1l_mu  = (const float*)d_in[11];
  const float* W1r_mu  = (const float*)d_in[12];
  const float* b1_mu   = (const float*)d_in[13];
  const float* W2l_um  = (const float*)d_in[14];
  const float* W2r_um  = (const float*)d_in[15];
  const float* b2_um   = (const float*)d_in[16];
  const float* W2l_mu  = (const float*)d_in[17];
  const float* W2r_mu  = (const float*)d_in[18];
  const float* b2_mu   = (const float*)d_in[19];
  const float* Wc1     = (const float*)d_in[20];
  const float* bc1     = (const float*)d_in[21];
  const float* Wc2     = (const float*)d_in[22];
  const float* bc2     = (const float*)d_in[23];

  // Workspace layout (floats). Accumulators first -> single zero pass.
  float* ws = (float*)d_ws;
  size_t off = 0;
  float* deg_m  = ws + off; off += NMer;
  float* deg_u  = ws + off; off += NUsr;
  float* agg1_m = ws + off; off += (size_t)NMer * FUin;
  float* agg1_u = ws + off; off += (size_t)NUsr * FMin;
  float* agg2_m = ws + off; off += (size_t)NMer * HH;
  float* agg2_u = ws + off; off += (size_t)NUsr * HH;
  const size_t zeroN = off;                 // 19,960,000 floats
  float* h_m = ws + off; off += (size_t)NMer * HH;
  float* h_u = ws + off; off += (size_t)NUsr * HH;
  float* z_m = ws + off; off += (size_t)NMer * HH;
  float* z_u = ws + off; off += (size_t)NUsr * HH;

  const long n4 = (long)(zeroN / 4);
  zero_kernel<<<(unsigned)((n4 + 255) / 256), 256, 0, stream>>>((float4*)ws, n4);

  // Degrees
  degree_kernel<<<(NEdg + 255) / 256, 256, 0, stream>>>(col_um, deg_m, NEdg);
  degree_kernel<<<(NEdg + 255) / 256, 256, 0, stream>>>(col_mu, deg_u, NEdg);

  // Layer-1 aggregation
  scatter_kernel<<<(unsigned)(((long)NEdg * (FUin / 4) + 255) / 256), 256, 0, stream>>>(
      x_user, row_um, col_um, agg1_m, NEdg, FUin);
  scatter_kernel<<<(unsigned)(((long)NEdg * (FMin / 4) + 255) / 256), 256, 0, stream>>>(
      x_merch, row_mu, col_mu, agg1_u, NEdg, FMin);

  // Layer-1 GEMMs (relu)
  sage_gemm_kernel<FUin, FMin, 1><<<NMer / 16, 256, 0, stream>>>(
      agg1_m, deg_m, x_merch, W1l_um, W1r_um, b1_um, h_m);
  sage_gemm_kernel<FMin, FUin, 1><<<NUsr / 16, 256, 0, stream>>>(
      agg1_u, deg_u, x_user, W1l_mu, W1r_mu, b1_mu, h_u);

  // Layer-2 aggregation
  scatter_kernel<<<(unsigned)(((long)NEdg * (HH / 4) + 255) / 256), 256, 0, stream>>>(
      h_u, row_um, col_um, agg2_m, NEdg, HH);
  scatter_kernel<<<(unsigned)(((long)NEdg * (HH / 4) + 255) / 256), 256, 0, stream>>>(
      h_m, row_mu, col_mu, agg2_u, NEdg, HH);

  // Layer-2 GEMMs (no activation)
  sage_gemm_kernel<HH, HH, 0><<<NMer / 16, 256, 0, stream>>>(
      agg2_m, deg_m, h_m, W2l_um, W2r_um, b2_um, z_m);
  sage_gemm_kernel<HH, HH, 0><<<NUsr / 16, 256, 0, stream>>>(
      agg2_u, deg_u, h_u, W2l_mu, W2r_mu, b2_mu, z_u);

  // Edge classifier
  edge_mlp_kernel<<<NTgt / 16, 256, 0, stream>>>(z_u, z_m, row_t, col_t,
                                                 Wc1, bc1, Wc2, bc2, (float*)d_out);
}